// SimpleTM_6116033429510
// MI455X (gfx1250) — compile-verified
//
#include <hip/hip_runtime.h>
#include <math.h>

// Problem constants (from reference)
#define N_LOCS 16384
#define N_REP  64
#define MAX_M  30
#define KPAD   32            // K padded to 32 (cols 30,31 zero)
#define XSTRIDE 33           // LDS padding: odd stride -> conflict-free column access
#define GSTRIDE 65
#define BLOCK  128           // 4 wave32 per location

typedef __attribute__((ext_vector_type(2))) float v2f;
typedef __attribute__((ext_vector_type(8))) float v8f;

__launch_bounds__(BLOCK)
__global__ void simpletm_kernel(const float* __restrict__ locs,
                                const float* __restrict__ response,
                                const int*   __restrict__ csets,
                                const float* __restrict__ nugget_params,
                                const float* __restrict__ theta_q,
                                const float* __restrict__ sigma_params,
                                const float* __restrict__ lengthscale,
                                float* __restrict__ part)   // per-location loglik
{
    __shared__ float X[N_REP * XSTRIDE];   // 64 x 33  (scaled neighbor matrix, K padded)
    __shared__ float G[N_REP * GSTRIDE];   // 64 x 65  (lower triangle: kernel -> Cholesky L)
    __shared__ float diagS[N_REP];         // S_ii = sum_k X[i][k]^2
    __shared__ float t[N_REP];             // y -> y_tilde (forward solve)
    __shared__ float sc[KPAD];             // per-k scaling sqrt(exp(-k*e^theta))
    __shared__ int   cidx[MAX_M];          // conditioning set for this location
    __shared__ float s_sc[8];              // 0:nug_mean 1:sig^2 2:1/ls^2 3:c 4:logdet 5:e^theta 6:1/Lkk
    __shared__ float red[BLOCK];

    const int b    = blockIdx.x;
    const int tid  = threadIdx.x;
    const int lane = tid & 31;
    const int wave = tid >> 5;

    // ---- scalar setup (thread 0) ----
    if (tid == 0) {
        const float np0 = nugget_params[0], np1 = nugget_params[1];
        const float sp0 = sigma_params[0],  sp1 = sigma_params[1];
        const float tq  = theta_q[0];
        const float ls  = lengthscale[0];
        // scales: d0[b] with scales[0] = d0[1]
        const int bb = (b == 0) ? 1 : b;
        int g0 = csets[bb * MAX_M + 0]; if (g0 < 0) g0 = 0;
        const float dx = locs[2 * bb]     - locs[2 * g0];
        const float dy = locs[2 * bb + 1] - locs[2 * g0 + 1];
        const float scale = sqrtf(dx * dx + dy * dy);
        const float lsc = logf(scale);
        float nm = expf(np0 + np1 * lsc);
        nm = fmaxf(nm - 1e-5f, 0.0f) + 1e-5f;            // relu(nm-1e-5)+1e-5
        const float sig = expf(sp0 + sp1 * lsc);
        const float c = sqrtf(2.0f * 1.5f);               // sqrt(2*SMOOTH)
        const float len_scal = expf(ls) * c;
        s_sc[0] = nm;
        s_sc[1] = sig * sig;
        s_sc[2] = 1.0f / (len_scal * len_scal);
        s_sc[3] = c;
        s_sc[4] = 0.0f;                                   // logdet accumulator
        s_sc[5] = expf(tq);
    }
    if (tid < MAX_M) cidx[tid] = csets[b * MAX_M + tid];
    __syncthreads();

    if (tid < KPAD) {
        const float eq = s_sc[5];
        sc[tid] = (tid < MAX_M) ? expf(-0.5f * (float)(tid + 1) * eq) : 0.0f;
    }
    __syncthreads();

    // ---- gather X into LDS: row i = replicate, col k = neighbor ----
    for (int idx = tid; idx < N_REP * KPAD; idx += BLOCK) {
        const int i = idx >> 5;       // replicate 0..63
        const int k = idx & 31;       // neighbor 0..31
        float v = 0.0f;
        if (k < MAX_M) {
            const int cs = cidx[k];
            if (cs >= 0) v = response[i * N_LOCS + cs] * sc[k];
        }
        X[i * XSTRIDE + k] = v;
    }
    if (tid < N_REP) t[tid] = response[tid * N_LOCS + b];   // y
    __syncthreads();

    // ---- diagonal of S (needed for the distance matrix) ----
    if (tid < N_REP) {
        float s = 0.0f;
        #pragma unroll
        for (int k = 0; k < KPAD; ++k) { const float v = X[tid * XSTRIDE + k]; s += v * v; }
        diagS[tid] = s;
    }
    __syncthreads();

    // ---- S = X X^T via f32 WMMA (16x16x4), lower-triangle tiles only ----
    // Cholesky reads only the lower triangle, and S is symmetric: 10 of 16 tiles.
    const float sig2    = s_sc[1];
    const float inv_ls2 = s_sc[2];
    const float c       = s_sc[3];
    const float inv_nm  = 1.0f / s_sc[0];

    const int m16 = lane & 15;
    const int hi  = lane >> 4;       // 0: K pair base 0 ; 1: base 2

    static const int ttr[10] = {0, 1, 1, 2, 2, 2, 3, 3, 3, 3};
    static const int ttc[10] = {0, 0, 1, 0, 1, 2, 0, 1, 2, 3};

    for (int ti = wave; ti < 10; ti += 4) {      // round-robin tiles over 4 waves
        const int tr = ttr[ti];
        const int tc = ttc[ti];
        v8f acc = {};
        #pragma unroll
        for (int ks = 0; ks < 8; ++ks) {
            const int k0 = ks * 4;
            // A: 16x4 slice of X rows tr*16..+15 ; VGPR0=K(0/2), VGPR1=K(1/3)
            v2f a, bf;
            const int ar = tr * 16 + m16;
            const int br = tc * 16 + m16;        // B[k][n] = X[n][k] (S symmetric)
            a.x  = X[ar * XSTRIDE + k0 + 2 * hi];
            a.y  = X[ar * XSTRIDE + k0 + 2 * hi + 1];
            bf.x = X[br * XSTRIDE + k0 + 2 * hi];
            bf.y = X[br * XSTRIDE + k0 + 2 * hi + 1];
            acc = __builtin_amdgcn_wmma_f32_16x16x4_f32(
                      false, a, false, bf, (short)0, acc, false, false);
        }
        // fused Matern transform in-register, store lower-triangle G tile
        #pragma unroll
        for (int r = 0; r < 8; ++r) {
            const int gi = tr * 16 + r + 8 * hi;    // C/D layout: VGPR r -> M = r + 8*hi
            const int gj = tc * 16 + m16;
            const float S = acc[r];
            const float d2   = (diagS[gi] + diagS[gj] - 2.0f * S) * inv_ls2;
            const float dist = __builtin_amdgcn_sqrtf(fmaxf(d2, 1e-12f));
            const float cd   = c * dist;
            const float nonlin = (1.0f + cd) * __expf(-cd);
            float g = (S + sig2 * nonlin) * inv_nm;
            if (gi == gj) g += 1.0f;
            G[gi * GSTRIDE + gj] = g;
        }
    }
    __syncthreads();

    // ---- Cholesky (right-looking, lower triangle) with fused forward substitution ----
    const int rown = tid & 63;       // candidate row offset
    const int jpar = tid >> 6;       // column-range split (0/1)
    for (int k = 0; k < N_REP; ++k) {
        if (tid == 0) {
            const float Lkk = sqrtf(G[k * GSTRIDE + k]);
            G[k * GSTRIDE + k] = Lkk;
            s_sc[4] += logf(Lkk);
            const float inv = 1.0f / Lkk;
            s_sc[6] = inv;
            t[k] *= inv;
        }
        __syncthreads();
        const float invL = s_sc[6];
        if (jpar == 0 && rown > k) G[rown * GSTRIDE + k] *= invL;
        __syncthreads();
        const int i = k + 1 + rown;              // this thread's trailing row
        const float tk = t[k];
        if (i < N_REP) {
            const float lik = G[i * GSTRIDE + k];
            // div-free split of columns k+1..i between the two thread halves
            for (int j = k + 1 + jpar; j <= i; j += 2)
                G[i * GSTRIDE + j] -= lik * G[j * GSTRIDE + k];
            if (jpar == 0) t[i] -= lik * tk;     // fused forward substitution
        }
        __syncthreads();
    }

    // ---- sum t^2 (block reduction) ----
    red[tid] = (tid < N_REP) ? t[tid] * t[tid] : 0.0f;
    __syncthreads();
    for (int s = BLOCK / 2; s > 0; s >>= 1) {
        if (tid < s) red[tid] += red[tid + s];
        __syncthreads();
    }

    if (tid == 0) {
        const float nmv     = s_sc[0];
        const float logdet  = s_sc[4];
        const float nug_sd  = nmv * 4.0f;                 // NUG_MULT
        const float alpha   = nmv * nmv / (nug_sd * nug_sd) + 2.0f;   // = 2.0625
        const float beta    = nmv * (alpha - 1.0f);
        const float alpha_p = alpha + (float)N_REP * 0.5f;
        const float beta_p  = beta + 0.5f * red[0];
        const float loglik  = -logdet + alpha * logf(beta) - alpha_p * logf(beta_p)
                              + lgammaf(alpha_p) - lgammaf(alpha);
        part[b] = loglik;
    }
}

// Deterministic fixed-order reduction of per-location logliks -> scalar output.
__launch_bounds__(256)
__global__ void reduce_kernel(const float* __restrict__ part, float* __restrict__ out)
{
    __shared__ float sm[256];
    float s = 0.0f;
    for (int i = threadIdx.x; i < N_LOCS; i += 256) s += part[i];
    sm[threadIdx.x] = s;
    __syncthreads();
    for (int st = 128; st > 0; st >>= 1) {
        if (threadIdx.x < st) sm[threadIdx.x] += sm[threadIdx.x + st];
        __syncthreads();
    }
    if (threadIdx.x == 0) out[0] = -sm[0];
}

extern "C" void kernel_launch(void* const* d_in, const int* in_sizes, int n_in,
                              void* d_out, int out_size, void* d_ws, size_t ws_size,
                              hipStream_t stream)
{
    (void)in_sizes; (void)n_in; (void)out_size; (void)ws_size;
    const float* locs     = (const float*)d_in[0];
    const float* response = (const float*)d_in[1];
    const int*   csets    = (const int*)  d_in[2];
    const float* nug      = (const float*)d_in[3];
    const float* tq       = (const float*)d_in[4];
    const float* sig      = (const float*)d_in[5];
    const float* ls       = (const float*)d_in[6];
    float* out  = (float*)d_out;
    float* part = (float*)d_ws;                 // 16384 floats = 64 KB scratch

    simpletm_kernel<<<N_LOCS, BLOCK, 0, stream>>>(locs, response, csets,
                                                  nug, tq, sig, ls, part);
    reduce_kernel<<<1, 256, 0, stream>>>(part, out);
}